// MultiScaleRoIAlign_15719580304179
// MI455X (gfx1250) — compile-verified
//
#include <hip/hip_runtime.h>
#include <stdint.h>

#define ROI_OUT 7
#define NBINS   49          // 7*7
#define C_CH    256
#define L_BOX   1024

#if defined(__HIP_DEVICE_COMPILE__) && defined(__gfx1250__) && \
    __has_builtin(__builtin_amdgcn_global_store_async_from_lds_b128) && \
    __has_builtin(__builtin_amdgcn_global_store_async_from_lds_b32)
#define USE_ASYNC_STORE 1
#else
#define USE_ASYNC_STORE 0
#endif

#if USE_ASYNC_STORE
// Exact pointee types per hipcc's diagnostic:
//   b128: 'vector_size(4*sizeof(int)) int __device__ *'
//   b32 : 'int __device__ *'
typedef int v4i32 __attribute__((vector_size(16)));
typedef __attribute__((address_space(1))) v4i32* gptr_v4i;  // global
typedef __attribute__((address_space(3))) v4i32* lptr_v4i;  // LDS
typedef __attribute__((address_space(1))) int*   gptr_i32;
typedef __attribute__((address_space(3))) int*   lptr_i32;
#endif

// One block per ROI. blockDim.x == 256 (one channel per thread, 8 wave32s).
__global__ __launch_bounds__(256)
void msroialign_kernel(const float* __restrict__ f0,
                       const float* __restrict__ f1,
                       const float* __restrict__ f2,
                       const float* __restrict__ f3,
                       const float* __restrict__ boxes,
                       float* __restrict__ out)
{
  __shared__ __align__(16) float  s_out[C_CH * NBINS];  // 50176 B staging tile
  __shared__ float4 s_w[NBINS];                         // bilinear weights (mask folded)
  __shared__ int    s_off[NBINS];                       // y0*W + x0

  const int roi = blockIdx.x;          // 0..R-1
  const int tid = threadIdx.x;         // 0..255
  const int n   = roi / L_BOX;         // batch index

  // Box (uniform across block; scalarized by the compiler)
  const float bx1 = boxes[roi * 4 + 0];
  const float by1 = boxes[roi * 4 + 1];
  const float bx2 = boxes[roi * 4 + 2];
  const float by2 = boxes[roi * 4 + 3];

  // FPN level selection: lvl = clip(floor(4 + log2(sqrt(area)/224 + 1e-8)), 2, 5) - 2
  const float area = fmaxf(bx2 - bx1, 0.0f) * fmaxf(by2 - by1, 0.0f);
  float lv = floorf(4.0f + log2f(sqrtf(area) * (1.0f / 224.0f) + 1e-8f));
  lv = fminf(fmaxf(lv, 2.0f), 5.0f);
  const int lvl = (int)lv - 2;

  const float* feat;
  int Hdim;
  switch (lvl) {                       // strides 4,8,16,32 over a 1024 image
    case 0:  feat = f0; Hdim = 256; break;
    case 1:  feat = f1; Hdim = 128; break;
    case 2:  feat = f2; Hdim = 64;  break;
    default: feat = f3; Hdim = 32;  break;
  }
  const int   Wdim  = Hdim;
  const float scale = (float)Hdim * (1.0f / 1024.0f);   // exact: 1/stride

  // ---- per-bin geometry, computed once by threads 0..48 --------------------
  if (tid < NBINS) {
    const int by = tid / ROI_OUT;
    const int bx = tid - by * ROI_OUT;

    const float x1 = bx1 * scale, y1 = by1 * scale;
    const float x2 = bx2 * scale, y2 = by2 * scale;
    const float roi_w = fmaxf(x2 - x1, 1.0f);
    const float roi_h = fmaxf(y2 - y1, 1.0f);
    const float bin_w = roi_w / 7.0f;
    const float bin_h = roi_h / 7.0f;

    const float ysf = y1 + ((float)by + 0.5f) * bin_h;
    const float xsf = x1 + ((float)bx + 0.5f) * bin_w;
    const bool  vy  = (ysf > -1.0f) && (ysf < (float)Hdim);
    const bool  vx  = (xsf > -1.0f) && (xsf < (float)Wdim);

    const float yc = fminf(fmaxf(ysf, 0.0f), (float)(Hdim - 1));
    const float xc = fminf(fmaxf(xsf, 0.0f), (float)(Wdim - 1));
    int y0 = (int)floorf(yc); y0 = (y0 < 0) ? 0 : ((y0 > Hdim - 2) ? Hdim - 2 : y0);
    int x0 = (int)floorf(xc); x0 = (x0 < 0) ? 0 : ((x0 > Wdim - 2) ? Wdim - 2 : x0);
    const float ly = yc - (float)y0;
    const float lx = xc - (float)x0;

    const float m   = (vy && vx) ? 1.0f : 0.0f;   // fold mask into weights
    const float wy0 = 1.0f - ly, wx0 = 1.0f - lx;
    s_w[tid]   = make_float4(m * wy0 * wx0, m * wy0 * lx,
                             m * ly  * wx0, m * ly  * lx);
    s_off[tid] = y0 * Wdim + x0;
  }
  __syncthreads();

  // ---- gather + bilinear: thread == channel -------------------------------
  const float* base = feat + (size_t)(n * C_CH + tid) * (size_t)(Hdim * Wdim);
  float* sc = s_out + tid * NBINS;     // lane stride 49: gcd(49,64)=1 -> no bank conflicts
  #pragma unroll 7
  for (int b = 0; b < NBINS; ++b) {
    const float4 w   = s_w[b];         // ds_read_b128 broadcast
    const int    off = s_off[b];
    const float* p   = base + off;
    const float v00 = p[0];
    const float v01 = p[1];
    const float v10 = p[Wdim];
    const float v11 = p[Wdim + 1];
    sc[b] = w.x * v00 + w.y * v01 + w.z * v10 + w.w * v11;
  }
  __syncthreads();

  // ---- coalesced drain of the 50KB LDS tile to global ---------------------
  float* gout = out + (size_t)roi * (C_CH * NBINS);
#if USE_ASYNC_STORE
  #pragma unroll
  for (int i = 0; i < 12; ++i) {       // 12 * 1024 floats via b128 (16B/lane)
    const int idx = i * 1024 + tid * 4;
    __builtin_amdgcn_global_store_async_from_lds_b128(
        (gptr_v4i)(uintptr_t)(gout + idx),
        (lptr_v4i)(uint32_t)(uintptr_t)(s_out + idx), 0, 0);
  }
  {                                    // tail: 256 floats via b32
    const int idx = 12 * 1024 + tid;
    __builtin_amdgcn_global_store_async_from_lds_b32(
        (gptr_i32)(uintptr_t)(gout + idx),
        (lptr_i32)(uint32_t)(uintptr_t)(s_out + idx), 0, 0);
  }
  #if __has_builtin(__builtin_amdgcn_s_wait_asynccnt)
    __builtin_amdgcn_s_wait_asynccnt(0);
  #else
    asm volatile("s_wait_asynccnt 0" ::: "memory");
  #endif
#else
  #pragma unroll
  for (int i = 0; i < 12; ++i) {
    const int idx = i * 1024 + tid * 4;
    *(float4*)(gout + idx) = *(const float4*)(s_out + idx);
  }
  gout[12 * 1024 + tid] = s_out[12 * 1024 + tid];
#endif
}

extern "C" void kernel_launch(void* const* d_in, const int* in_sizes, int n_in,
                              void* d_out, int out_size, void* d_ws, size_t ws_size,
                              hipStream_t stream) {
  const float* f0    = (const float*)d_in[0];
  const float* f1    = (const float*)d_in[1];
  const float* f2    = (const float*)d_in[2];
  const float* f3    = (const float*)d_in[3];
  const float* boxes = (const float*)d_in[4];
  float*       out   = (float*)d_out;

  const int R = in_sizes[4] / 4;       // N * L = 2048 ROIs
  msroialign_kernel<<<R, 256, 0, stream>>>(f0, f1, f2, f3, boxes, out);
}